// spatial_76940044141091
// MI455X (gfx1250) — compile-verified
//
#include <hip/hip_runtime.h>

#ifndef __has_builtin
#define __has_builtin(x) 0
#endif

// Problem constants (reference: x = f32[32,128,192,192], 3x3 stride-3 sum-pool mask)
#define POOL 3
#define WW   192                       // row width (floats)
#define TRIPLES 16                     // row-triples per block tile
#define ROWS (TRIPLES * POOL)          // 48 rows per tile
#define TILE_FLOATS (ROWS * WW)        // 9216 floats = 36 KB
#define NTHREADS 256
#define CELLS (TRIPLES * (WW / POOL))           // 1024 pooled 3x3 cells per tile
#define CELLS_PER_THREAD (CELLS / NTHREADS)     // 4 cells per thread

struct __align__(4) f3 { float x, y, z; };  // 12-byte store unit

// One async DMA copy: 16 B per lane, 512 B per wave instruction. INST_OFFSET is
// applied to BOTH the LDS destination and the global source address (ISA
// async-addressing rule), so all copies share one (lds, global) address pair.
#define ASYNC_CP16(OFF)                                                        \
    asm volatile("global_load_async_to_lds_b128 %0, %1, off offset:" #OFF      \
                 " th:TH_LOAD_NT"                                              \
                 :: "v"(loff), "v"(gaddr) : "memory")

__global__ __launch_bounds__(NTHREADS)
void pool_mask_upsample(const float* __restrict__ x, float* __restrict__ out)
{
    __shared__ float tile[TILE_FLOATS];

    // Flat tile index: TILE_FLOATS divides a 192*192 plane exactly (4 tiles/plane),
    // so a tile never straddles a (b,c) plane and pooling windows stay intact.
    const long long base = (long long)blockIdx.x * (long long)TILE_FLOATS;
    const float* gsrc = x + base;
    float*       gdst = out + base;

    // LDS byte offset of tile[]: low 32 bits of the generic pointer (LDS aperture
    // maps addr[31:0] directly to the LDS byte address).
    const unsigned lds0 = (unsigned)(unsigned long long)(const void*)&tile[0];

    // ---- Stage 1: async global -> LDS DMA (ASYNCcnt path, no VGPR data) ----
    // 9216 floats / block: thread t copies 16 B at byte (t*16 + k*4096), k=0..8.
    {
        const unsigned loff = lds0 + (unsigned)threadIdx.x * 16u;
        const unsigned long long gaddr =
            (unsigned long long)(const void*)(gsrc + (size_t)threadIdx.x * 4);
        ASYNC_CP16(0);
        ASYNC_CP16(4096);
        ASYNC_CP16(8192);
        ASYNC_CP16(12288);
        ASYNC_CP16(16384);
        ASYNC_CP16(20480);
        ASYNC_CP16(24576);
        ASYNC_CP16(28672);
        ASYNC_CP16(32768);
    }

    // Wait for this wave's async copies, then sync the workgroup so every thread
    // can read data DMA'd by other waves.
#if __has_builtin(__builtin_amdgcn_s_wait_asynccnt)
    __builtin_amdgcn_s_wait_asynccnt(0);
#else
    asm volatile("s_wait_asynccnt 0x0" ::: "memory");
#endif
    __syncthreads();

    // ---- Stage 2: four pooled 3x3 cells per thread ----
    #pragma unroll
    for (int k = 0; k < CELLS_PER_THREAD; ++k) {
        const int cell = (int)threadIdx.x + k * NTHREADS;
        const int trip = cell >> 6;          // /64 pooled columns per row-triple
        const int col  = cell & 63;
        const int o0   = trip * (3 * WW) + col * 3;

        float v[9];
        #pragma unroll
        for (int r = 0; r < 3; ++r) {
            #pragma unroll
            for (int c = 0; c < 3; ++c)
                v[r * 3 + c] = tile[o0 + r * WW + c];
        }
        const float s = ((v[0] + v[1]) + (v[2] + v[3]))
                      + ((v[4] + v[5]) + (v[6] + v[7])) + v[8];
        const float m = (s > 0.0f) ? 1.0f : 0.0f;

        #pragma unroll
        for (int r = 0; r < 3; ++r) {
            f3 o;
            o.x = v[r * 3 + 0] * m;
            o.y = v[r * 3 + 1] * m;
            o.z = v[r * 3 + 2] * m;
            *reinterpret_cast<f3*>(gdst + o0 + r * WW) = o;   // 12B, 4B-aligned
        }
    }
}

extern "C" void kernel_launch(void* const* d_in, const int* in_sizes, int n_in,
                              void* d_out, int out_size, void* d_ws, size_t ws_size,
                              hipStream_t stream)
{
    (void)n_in; (void)out_size; (void)d_ws; (void)ws_size;
    const float* x   = (const float*)d_in[0];
    float*       out = (float*)d_out;

    const long long n = (long long)in_sizes[0];          // 32*128*192*192 = 150,994,944
    const int nblocks = (int)(n / TILE_FLOATS);          // 16384 tiles
    hipLaunchKernelGGL(pool_mask_upsample, dim3(nblocks), dim3(NTHREADS), 0, stream,
                       x, out);
}